// DiffusionOperator_279
// MI455X (gfx1250) — compile-verified
//
#include <hip/hip_runtime.h>
#include <math.h>

#define N_NODES   100000
#define N_EDGES   1600000
#define N_CH      64
#define HID       32
#define FAN1      (N_CH + 4)      // 68 = 17 * 4
#define NCOEF     6               // POLY_DEGREE + 1

typedef float v2f __attribute__((ext_vector_type(2)));
typedef float v8f __attribute__((ext_vector_type(8)));

// ---------------------------------------------------------------- utilities
__global__ void zero_kernel(float* __restrict__ p, size_t n) {
    size_t i = (size_t)blockIdx.x * blockDim.x + threadIdx.x;
    size_t stride = (size_t)gridDim.x * blockDim.x;
    for (; i < n; i += stride) p[i] = 0.0f;
}

// ---------------------------------------------------------------- degree
__global__ void deg_kernel(const int* __restrict__ col, float* __restrict__ deg, int E) {
    int e = blockIdx.x * blockDim.x + threadIdx.x;
    if (e < E) atomicAdd(&deg[col[e]], 1.0f);      // global_atomic_add_f32 (no return)
}

// dis = min(deg^-0.5, 1e6)  (deg==0 -> rsqrt=inf -> clamped to 1e6, matching ref)
__global__ void dis_kernel(const float* __restrict__ deg, float* __restrict__ dis, int n) {
    int i = blockIdx.x * blockDim.x + threadIdx.x;
    if (i < n) dis[i] = fminf(rsqrtf(deg[i]), 1.0e6f);
}

// ---------------------------------------------------------------- global stats
// stats[0..63] = per-channel sums, stats[64] = sum of squares
__global__ __launch_bounds__(256)
void stats_kernel(const float* __restrict__ x, float* __restrict__ stats, size_t n) {
    __shared__ float chs[N_CH];
    __shared__ float ssq;
    if (threadIdx.x < N_CH) chs[threadIdx.x] = 0.0f;
    if (threadIdx.x == 0)  ssq = 0.0f;
    __syncthreads();
    float lsq = 0.0f;
    size_t i = (size_t)blockIdx.x * blockDim.x + threadIdx.x;
    size_t stride = (size_t)gridDim.x * blockDim.x;
    for (; i < n; i += stride) {
        float v = x[i];
        lsq += v * v;
        atomicAdd(&chs[i & (N_CH - 1)], v);        // ds_add_f32
    }
    atomicAdd(&ssq, lsq);
    __syncthreads();
    if (threadIdx.x < N_CH) atomicAdd(&stats[threadIdx.x], chs[threadIdx.x]);
    if (threadIdx.x == 0)   atomicAdd(&stats[N_CH], ssq);
}

// ---------------------------------------------------------------- coefficient MLP
// Single wave32, EXEC all-ones: layer-1 (32x68 @ 68-vec) via chained
// V_WMMA_F32_16X16X4_F32 accumulation; B-fragment broadcasts the input
// vector across all 16 N-columns, so every column of D equals W1 @ v.
__global__ __launch_bounds__(32)
void mlp_kernel(const float* __restrict__ stats,
                const float* __restrict__ W1, const float* __restrict__ b1,
                const float* __restrict__ W2, const float* __restrict__ b2,
                float* __restrict__ coeffs) {
    __shared__ float cin[FAN1];
    __shared__ float h[HID];
    __shared__ float o6[NCOEF];
    const int l = threadIdx.x;           // 0..31
    const int half = l >> 4;             // lane half (0: lanes 0-15, 1: lanes 16-31)
    const int m = l & 15;

    // channel means
    cin[l]      = stats[l]      * (1.0f / (float)N_NODES);
    cin[l + 32] = stats[l + 32] * (1.0f / (float)N_NODES);
    __syncthreads();
    if (l == 0) {
        float tot = 0.0f;
        for (int i = 0; i < N_CH; ++i) tot += stats[i];
        const float nc = (float)N_NODES * (float)N_CH;
        float mean = tot / nc;
        float var  = (stats[N_CH] - tot * tot / nc) / (nc - 1.0f);  // ddof=1
        cin[64] = mean;
        cin[65] = sqrtf(fmaxf(var, 0.0f));
        cin[66] = (float)N_NODES;
        cin[67] = (float)N_EDGES;
    }
    __syncthreads();

#if defined(__gfx1250__) && __has_builtin(__builtin_amdgcn_wmma_f32_16x16x4_f32)
    v8f acc0 = {0.f,0.f,0.f,0.f,0.f,0.f,0.f,0.f};
    v8f acc1 = {0.f,0.f,0.f,0.f,0.f,0.f,0.f,0.f};
    for (int kb = 0; kb < FAN1 / 4; ++kb) {          // 17 K-chunks of 4
        const int k0 = kb * 4 + half * 2;
        // A (16x4 f32): lane l, VGPR t holds A[l%16][ (l/16)*2 + t ]
        v2f a0, a1, bf;
        a0.x = W1[m * FAN1 + k0];          a0.y = W1[m * FAN1 + k0 + 1];
        a1.x = W1[(m + 16) * FAN1 + k0];   a1.y = W1[(m + 16) * FAN1 + k0 + 1];
        // B (4x16 f32): broadcast v over all N columns; K index mirrors A
        bf.x = cin[k0];                    bf.y = cin[k0 + 1];
        acc0 = __builtin_amdgcn_wmma_f32_16x16x4_f32(
                   false, a0, false, bf, (short)0, acc0, false, false);
        acc1 = __builtin_amdgcn_wmma_f32_16x16x4_f32(
                   false, a1, false, bf, (short)0, acc1, false, false);
    }
    // D layout: VGPR t -> row t (lanes 0-15) / row t+8 (lanes 16-31)
    for (int t = 0; t < 8; ++t) {
        int r = t + half * 8;
        h[r]      = fmaxf(acc0[t] + b1[r],      0.0f);  // redundant same-value writes: benign
        h[r + 16] = fmaxf(acc1[t] + b1[r + 16], 0.0f);
    }
#else
    if (l < HID) {
        float s = b1[l];
        for (int i = 0; i < FAN1; ++i) s += W1[l * FAN1 + i] * cin[i];
        h[l] = fmaxf(s, 0.0f);
    }
#endif
    __syncthreads();

    if (l < NCOEF) {
        float s = b2[l];
        for (int j = 0; j < HID; ++j) s += W2[l * HID + j] * h[j];
        o6[l] = s;
    }
    __syncthreads();
    if (l == 0) {
        float mx = o6[0];
        for (int k = 1; k < NCOEF; ++k) mx = fmaxf(mx, o6[k]);
        float ex[NCOEF], den = 0.0f;
        for (int k = 0; k < NCOEF; ++k) { ex[k] = expf(o6[k] - mx); den += ex[k]; }
        float inv = 1.0f / den;
        for (int k = 0; k < NCOEF; ++k) coeffs[k] = ex[k] * inv;
    }
}

// ---------------------------------------------------------------- result = c0 * x
__global__ void init_kernel(float* __restrict__ res, const float* __restrict__ x,
                            const float* __restrict__ coeffs, size_t n) {
    float c0 = coeffs[0];
    size_t i = (size_t)blockIdx.x * blockDim.x + threadIdx.x;
    size_t stride = (size_t)gridDim.x * blockDim.x;
    for (; i < n; i += stride) res[i] = c0 * x[i];
}

// ---------------------------------------------------------------- SpMM (fused norm)
// One wave32 per edge: row/col/norm are wave-uniform; each lane moves 2
// channels (coalesced float2 gather + 2 f32 atomic scatters). HBM-bound.
__global__ __launch_bounds__(256)
void spmm_kernel(const int* __restrict__ rowi, const int* __restrict__ coli,
                 const float* __restrict__ dis,
                 const float* __restrict__ tx, float* __restrict__ out, int E) {
    long long gid = (long long)blockIdx.x * blockDim.x + threadIdx.x;
    int e = (int)(gid >> 5);
    if (e >= E) return;
    int lane = (int)(gid & 31);
    int r = rowi[e], c = coli[e];
    float nrm = dis[r] * dis[c];
    const float2* src = reinterpret_cast<const float2*>(tx + (size_t)c * N_CH);
    float2 v = src[lane];
    float* dst = out + (size_t)r * N_CH + lane * 2;
    atomicAdd(dst,     nrm * v.x);
    atomicAdd(dst + 1, nrm * v.y);
}

// ---------------------------------------------------------------- res += ck*cur ; other = 0
__global__ void axpy_zero_kernel(float* __restrict__ res, const float* __restrict__ cur,
                                 float* __restrict__ other,
                                 const float* __restrict__ coeffs, int k, size_t n) {
    float ck = coeffs[k];
    size_t i = (size_t)blockIdx.x * blockDim.x + threadIdx.x;
    size_t stride = (size_t)gridDim.x * blockDim.x;
    for (; i < n; i += stride) {
        res[i] += ck * cur[i];
        if (other) other[i] = 0.0f;
    }
}

// ---------------------------------------------------------------- driver
extern "C" void kernel_launch(void* const* d_in, const int* in_sizes, int n_in,
                              void* d_out, int out_size, void* d_ws, size_t ws_size,
                              hipStream_t stream) {
    const float* x  = (const float*)d_in[0];
    const int*   ei = (const int*)  d_in[1];
    const float* W1 = (const float*)d_in[2];
    const float* b1 = (const float*)d_in[3];
    const float* W2 = (const float*)d_in[4];
    const float* b2 = (const float*)d_in[5];
    float* out = (float*)d_out;

    const int* rowi = ei;
    const int* coli = ei + N_EDGES;

    const size_t NC = (size_t)N_NODES * N_CH;
    float* ws     = (float*)d_ws;
    float* deg    = ws;                       // N
    float* dis    = deg + N_NODES;            // N
    float* stats  = dis + N_NODES;            // 72
    float* coeffs = stats + 72;               // 8
    float* bufA   = coeffs + 8;               // N*C
    float* bufB   = bufA + NC;                // N*C   (~52 MB total)

    const int spmm_blocks = (int)(((long long)N_EDGES * 32 + 255) / 256);  // 200000

    zero_kernel<<<512, 256, 0, stream>>>(ws, 2 * (size_t)N_NODES + 80);
    zero_kernel<<<2048, 256, 0, stream>>>(bufA, NC);

    deg_kernel<<<(N_EDGES + 255) / 256, 256, 0, stream>>>(coli, deg, N_EDGES);
    dis_kernel<<<(N_NODES + 255) / 256, 256, 0, stream>>>(deg, dis, N_NODES);
    stats_kernel<<<1024, 256, 0, stream>>>(x, stats, NC);
    mlp_kernel<<<1, 32, 0, stream>>>(stats, W1, b1, W2, b2, coeffs);

    init_kernel<<<2048, 256, 0, stream>>>(out, x, coeffs, NC);

    // k=1..5 ping-pong; each destination is zero before its SpMM.
    spmm_kernel<<<spmm_blocks, 256, 0, stream>>>(rowi, coli, dis, x,    bufA, N_EDGES);
    axpy_zero_kernel<<<2048, 256, 0, stream>>>(out, bufA, bufB, coeffs, 1, NC);
    spmm_kernel<<<spmm_blocks, 256, 0, stream>>>(rowi, coli, dis, bufA, bufB, N_EDGES);
    axpy_zero_kernel<<<2048, 256, 0, stream>>>(out, bufB, bufA, coeffs, 2, NC);
    spmm_kernel<<<spmm_blocks, 256, 0, stream>>>(rowi, coli, dis, bufB, bufA, N_EDGES);
    axpy_zero_kernel<<<2048, 256, 0, stream>>>(out, bufA, bufB, coeffs, 3, NC);
    spmm_kernel<<<spmm_blocks, 256, 0, stream>>>(rowi, coli, dis, bufA, bufB, N_EDGES);
    axpy_zero_kernel<<<2048, 256, 0, stream>>>(out, bufB, bufA, coeffs, 4, NC);
    spmm_kernel<<<spmm_blocks, 256, 0, stream>>>(rowi, coli, dis, bufB, bufA, N_EDGES);
    axpy_zero_kernel<<<2048, 256, 0, stream>>>(out, bufA, nullptr, coeffs, 5, NC);
}